// HausdorffDTLoss_80745385165232
// MI455X (gfx1250) — compile-verified
//
#include <hip/hip_runtime.h>
#include <hip/hip_bf16.h>

// ---------------------------------------------------------------------------
// HausdorffDTLoss for MI455X (gfx1250, wave32).
//   logits  : f32 [16,1,256,256]
//   targets : i32 [16,1,256,256]
//   out     : f32 scalar
// Workspace: 4 EDT^2 planes (16 MB) + 256 block partials.
// ---------------------------------------------------------------------------

#define B_DIM 16
#define H_DIM 256
#define W_DIM 256
#define PLANE (B_DIM * H_DIM * W_DIM)   // 1048576
#define INF1  1.0e4f
#define INF2  1.0e8f                    // INF1^2, exact in f32

typedef __attribute__((ext_vector_type(2))) float        v2f;
typedef __attribute__((ext_vector_type(8))) float        v8f;
typedef __attribute__((ext_vector_type(4))) unsigned int v4u;
typedef __attribute__((ext_vector_type(4))) int          v4i;
typedef __attribute__((ext_vector_type(8))) int          v8i;

#define HAS_WMMA_F32X4 __has_builtin(__builtin_amdgcn_wmma_f32_16x16x4_f32)
#define HAS_TDM        (__has_builtin(__builtin_amdgcn_tensor_load_to_lds) && \
                        __has_builtin(__builtin_amdgcn_s_wait_tensorcnt))

// Single-instruction xor-lane add: ds_swizzle_b32 group-of-32 mode,
// src_lane = ((lane & 0x1f) | 0) ^ mask.
#define SWZ_XOR_ADD(x, m)                                                      \
    ((x) + __int_as_float(__builtin_amdgcn_ds_swizzle(__float_as_int(x),       \
                                                      (((m) << 10) | 0x1f))))

// Sum 32 lane values of a wave. V_WMMA_F32_16X16X4_F32 with an all-ones A:
// D[m,n] = sum_k B[k,n] + C[m,n]. Lane partials in B vgpr0 give
// C[0,n] = p_n + p_{n+16}; all D rows are identical under ones-A, so every
// lane's d[0] holds the pair sum for lane%16. Four ds_swizzle xor folds
// finish the 16-value reduction (exact f32 throughout).
__device__ inline float wave_reduce_sum(float v) {
#if HAS_WMMA_F32X4
    v2f a; a.x = 1.0f; a.y = 1.0f;
    v2f b; b.x = v;    b.y = 0.0f;
    v8f c = {};
    c = __builtin_amdgcn_wmma_f32_16x16x4_f32(false, a, false, b,
                                              (short)0, c, false, false);
    float x = c[0];
    x = SWZ_XOR_ADD(x, 1);
    x = SWZ_XOR_ADD(x, 2);
    x = SWZ_XOR_ADD(x, 4);
    x = SWZ_XOR_ADD(x, 8);
    return x;
#else
    float x = v;
    x = SWZ_XOR_ADD(x, 16);
    x = SWZ_XOR_ADD(x, 8);
    x = SWZ_XOR_ADD(x, 4);
    x = SWZ_XOR_ADD(x, 2);
    x = SWZ_XOR_ADD(x, 1);
    return x;
#endif
}

// ---------------------------------------------------------------------------
// Kernel 1: per-row 1D pass. One thread per (b,h) row (4096 threads).
// Plane 0: edt^2(pred) 1D, 1: edt^2(~pred), 2: edt^2(tgt), 3: edt^2(~tgt).
// Matches the JAX scan exactly: counters start at 1e4, +1 per step, reset on
// background; final d = min(min(fwd,bwd), 1e4); store d*d.
// ---------------------------------------------------------------------------
__global__ void k1_rowpass(const float* __restrict__ logits,
                           const int*   __restrict__ targets,
                           float*       __restrict__ D) {
    const int r    = blockIdx.x * blockDim.x + threadIdx.x;  // 0..4095
    const int base = r << 8;                                 // r * 256

    unsigned pb[8], tb[8];
#pragma unroll
    for (int q = 0; q < 8; ++q) { pb[q] = 0u; tb[q] = 0u; }

    float c0 = INF1, c1 = INF1, c2 = INF1, c3 = INF1;
    for (int w = 0; w < W_DIM; ++w) {
        const float lg = logits[base + w];
        const int   tg = targets[base + w];
        const unsigned p  = (lg > 0.0f) ? 1u : 0u;   // sigmoid(x)>0.5 <=> x>0
        const unsigned tt = (tg != 0)   ? 1u : 0u;
        pb[w >> 5] |= p  << (w & 31);
        tb[w >> 5] |= tt << (w & 31);
        c0 = p  ? c0 + 1.0f : 0.0f;
        c1 = p  ? 0.0f : c1 + 1.0f;
        c2 = tt ? c2 + 1.0f : 0.0f;
        c3 = tt ? 0.0f : c3 + 1.0f;
        D[0 * PLANE + base + w] = c0;
        D[1 * PLANE + base + w] = c1;
        D[2 * PLANE + base + w] = c2;
        D[3 * PLANE + base + w] = c3;
    }

    c0 = c1 = c2 = c3 = INF1;
    for (int w = W_DIM - 1; w >= 0; --w) {
        const unsigned p  = (pb[w >> 5] >> (w & 31)) & 1u;
        const unsigned tt = (tb[w >> 5] >> (w & 31)) & 1u;
        c0 = p  ? c0 + 1.0f : 0.0f;
        c1 = p  ? 0.0f : c1 + 1.0f;
        c2 = tt ? c2 + 1.0f : 0.0f;
        c3 = tt ? 0.0f : c3 + 1.0f;
        float d0 = fminf(fminf(D[0 * PLANE + base + w], c0), INF1);
        float d1 = fminf(fminf(D[1 * PLANE + base + w], c1), INF1);
        float d2 = fminf(fminf(D[2 * PLANE + base + w], c2), INF1);
        float d3 = fminf(fminf(D[3 * PLANE + base + w], c3), INF1);
        D[0 * PLANE + base + w] = d0 * d0;
        D[1 * PLANE + base + w] = d1 * d1;
        D[2 * PLANE + base + w] = d2 * d2;
        D[3 * PLANE + base + w] = d3 * d3;
    }
}

// ---------------------------------------------------------------------------
// Kernel 2: column min-plus pass + select + partial reduction.
// Grid: 256 blocks = (b, wtile); 512 threads (16 waves).
// LDS tile: 4 planes x 256 rows x 16 cols f32 = 64 KB, filled by the Tensor
// Data Mover (one 2D-tile DMA per plane, issued by wave 0) when available.
// Thread t: column w = t&15, rows [(t>>4)*8, +8), 4 planes of register
// accumulators (32 VGPRs of state -> no spills).
// ---------------------------------------------------------------------------
__global__ void __launch_bounds__(512)
k2_colpass(const float* __restrict__ D,
           const float* __restrict__ logits,
           const int*   __restrict__ targets,
           float*       __restrict__ partials) {
    __shared__ float lds[4 * H_DIM * 16];   // 64 KB

    const int blk = blockIdx.x;
    const int b   = blk >> 4;         // batch
    const int wt  = blk & 15;         // 16-wide column tile index
    const int t   = threadIdx.x;

#if HAS_TDM
    // --- Tensor Data Mover tile fill: 2D tile 16 cols x 256 rows, 4B elems,
    // row stride 256 elems, packed contiguously into LDS (row = 64B). ---
    if (t < 32) {
#pragma unroll
        for (int p = 0; p < 4; ++p) {
            const unsigned long long ga =
                (unsigned long long)(const void*)(D + p * PLANE +
                                                  b * (H_DIM * W_DIM) + wt * 16);
            const unsigned ldsoff =
                (unsigned)(unsigned long long)(&lds[p * (H_DIM * 16)]);
            v4u g0;
            g0.x = 1u;                                   // count=1, user D#
            g0.y = ldsoff;                               // lds_addr (bytes)
            g0.z = (unsigned)(ga & 0xffffffffu);         // global_addr[31:0]
            g0.w = (unsigned)((ga >> 32) & 0x01ffffffu)  // global_addr[56:32]
                 | 0x80000000u;                          // type=2 ("image")
            v8i g1;
            g1[0] = 0x00020000;          // workgroup_mask=0, data_size=4B
            g1[1] = (int)(W_DIM << 16);  // tensor_dim0 = 256   (bits 79:48)
            g1[2] = (int)(H_DIM << 16);  // tensor_dim1 = 256   (bits 111:80 lo)
            g1[3] = (int)(16 << 16);     // tile_dim0  = 16     (bits 127:112)
            g1[4] = H_DIM;               // tile_dim1  = 256    (bits 143:128)
            g1[5] = W_DIM;               // tensor_dim0_stride = 256
            g1[6] = 0;
            g1[7] = 0;
            const v4i z4 = {0, 0, 0, 0};
#if __clang_major__ >= 23
            const v8i z8 = {0, 0, 0, 0, 0, 0, 0, 0};
            __builtin_amdgcn_tensor_load_to_lds(g0, g1, z4, z4, z8, 0);
#else
            __builtin_amdgcn_tensor_load_to_lds(g0, g1, z4, z4, 0);
#endif
        }
    }
    __builtin_amdgcn_s_wait_tensorcnt(0);   // issuing wave gates the barrier
    __syncthreads();
#else
    // --- Fallback fill: thread t loads half-row (j = t>>1) of each plane. ---
    {
        const int j    = t >> 1;
        const int half = (t & 1) * 8;
        const int rowbase = b * (H_DIM * W_DIM) + j * W_DIM + wt * 16 + half;
#pragma unroll
        for (int p = 0; p < 4; ++p) {
            const float4* src = (const float4*)(D + p * PLANE + rowbase);
            float4* dst = (float4*)(&lds[p * (H_DIM * 16) + j * 16 + half]);
            dst[0] = src[0];
            dst[1] = src[1];
        }
    }
    __syncthreads();
#endif

    const int w     = t & 15;
    const int ibase = (t >> 4) << 3;          // 8 rows per thread
    const int wglob = wt * 16 + w;

    // Warm the epilogue reads while the VALU loop runs.
    __builtin_prefetch(&logits[b * (H_DIM * W_DIM) + ibase * W_DIM + wglob], 0, 1);
    __builtin_prefetch(&targets[b * (H_DIM * W_DIM) + ibase * W_DIM + wglob], 0, 1);

    float acc0[8], acc1[8], acc2[8], acc3[8];
#pragma unroll
    for (int k = 0; k < 8; ++k) {
        acc0[k] = INF2; acc1[k] = INF2; acc2[k] = INF2; acc3[k] = INF2;
    }

    for (int j = 0; j < H_DIM; ++j) {
        const float r0 = lds[0 * (H_DIM * 16) + j * 16 + w];
        const float r1 = lds[1 * (H_DIM * 16) + j * 16 + w];
        const float r2 = lds[2 * (H_DIM * 16) + j * 16 + w];
        const float r3 = lds[3 * (H_DIM * 16) + j * 16 + w];
        const float f  = (float)(ibase - j);       // |.| <= 255, exact
#pragma unroll
        for (int k = 0; k < 8; ++k) {
            const float diff = f + (float)k;
            const float d2   = diff * diff;        // exact
            acc0[k] = fminf(acc0[k], d2 + r0);
            acc1[k] = fminf(acc1[k], d2 + r1);
            acc2[k] = fminf(acc2[k], d2 + r2);
            acc3[k] = fminf(acc3[k], d2 + r3);
        }
    }

    // Per-pixel select + ALPHA=2 (mimic reference's sqrt-then-square).
    float s = 0.0f;
#pragma unroll
    for (int k = 0; k < 8; ++k) {
        const int i   = ibase + k;
        const int idx = b * (H_DIM * W_DIM) + i * W_DIM + wglob;
        const bool pf = logits[idx] > 0.0f;
        const bool tf = targets[idx] != 0;
        float ap = tf ? acc1[k] : acc0[k];   // pred_dt^2 source
        float at = pf ? acc3[k] : acc2[k];   // target_dt^2 source
        ap = sqrtf(ap); at = sqrtf(at);
        s += ap * ap + at * at;
    }

    // Block reduction: WMMA wave fold, then 16 wave sums via reused LDS.
    const float ws = wave_reduce_sum(s);
    __syncthreads();                          // tile reads fully done
    if ((t & 31) == 0) lds[t >> 5] = ws;
    __syncthreads();
    if (t == 0) {
        float tot = 0.0f;
#pragma unroll
        for (int i = 0; i < 16; ++i) tot += lds[i];
        partials[blk] = tot;
    }
}

// ---------------------------------------------------------------------------
// Kernel 3: reduce 256 block partials with 4 chained f32 WMMAs (ones x B + C),
// fold 16 column sums with ds_swizzle, scale by 1/(2*N) = 2^-21 (exact).
// ---------------------------------------------------------------------------
__global__ void k3_final(const float* __restrict__ partials,
                         float*       __restrict__ out,
                         float scale) {
    const int lane = threadIdx.x;   // 32 threads, 1 wave
#if HAS_WMMA_F32X4
    v2f a; a.x = 1.0f; a.y = 1.0f;
    v8f c = {};
    for (int s = 0; s < 256; s += 64) {
        v2f bv;
        bv.x = partials[s + lane];
        bv.y = partials[s + 32 + lane];
        c = __builtin_amdgcn_wmma_f32_16x16x4_f32(false, a, false, bv,
                                                  (short)0, c, false, false);
    }
    float x = c[0];
    x = SWZ_XOR_ADD(x, 1);
    x = SWZ_XOR_ADD(x, 2);
    x = SWZ_XOR_ADD(x, 4);
    x = SWZ_XOR_ADD(x, 8);
    if (lane == 0) out[0] = x * scale;
#else
    if (lane == 0) {
        float tot = 0.0f;
        for (int i = 0; i < 256; ++i) tot += partials[i];
        out[0] = tot * scale;
    }
#endif
}

// ---------------------------------------------------------------------------
extern "C" void kernel_launch(void* const* d_in, const int* in_sizes, int n_in,
                              void* d_out, int out_size, void* d_ws, size_t ws_size,
                              hipStream_t stream) {
    (void)in_sizes; (void)n_in; (void)out_size; (void)ws_size;

    const float* logits  = (const float*)d_in[0];
    const int*   targets = (const int*)d_in[1];
    float*       out     = (float*)d_out;

    float* planes   = (float*)d_ws;            // 4 * PLANE floats = 16 MB
    float* partials = planes + 4 * PLANE;      // 256 floats

    // 1) row pass: 4096 rows, one thread each.
    k1_rowpass<<<16, 256, 0, stream>>>(logits, targets, planes);

    // 2) column pass + combine: (16 batches x 16 column tiles) blocks.
    k2_colpass<<<256, 512, 0, stream>>>(planes, logits, targets, partials);

    // 3) final WMMA reduction. scale = 1/(2*B*H*W) = 2^-21 exactly.
    const float scale = 1.0f / (2.0f * (float)PLANE);
    k3_final<<<1, 32, 0, stream>>>(partials, out, scale);
}